// Adder2d_24756191494450
// MI455X (gfx1250) — compile-verified
//
#include <hip/hip_runtime.h>
#include <stdint.h>

// Problem constants (from reference)
#define N_    16
#define C_    64
#define HW_   14
#define F_    64
#define KS    3
#define P_    (HW_ * HW_)     // 196
#define D_    (C_ * KS * KS)  // 576
#define FT    16              // filters per block
#define PADHW 16              // 14 + 2*pad
#define NTHREADS 256

#ifndef __has_builtin
#define __has_builtin(x) 0
#endif

#if __has_builtin(__builtin_amdgcn_global_load_async_to_lds_b32)
#define USE_ASYNC_BUILTIN 1
#endif
#if __has_builtin(__builtin_amdgcn_s_wait_asynccnt)
#define USE_WAITA_BUILTIN 1
#endif
#if __has_builtin(__builtin_amdgcn_tensor_load_to_lds)
#define USE_TDM 1
#endif
#if __has_builtin(__builtin_amdgcn_s_wait_tensorcnt)
#define USE_WAITT_BUILTIN 1
#endif

// Typed address-space pointers matching the async builtin prototypes
typedef int v4i __attribute__((vector_size(16)));
typedef __attribute__((address_space(1))) int  g_i32;
typedef __attribute__((address_space(3))) int  l_i32;
typedef __attribute__((address_space(1))) v4i  g_v4i;
typedef __attribute__((address_space(3))) v4i  l_v4i;
typedef __attribute__((address_space(3))) void as3_void;

// TDM descriptor vector types (this toolchain: 6-arg form
//   (u32x4 g0, i32x8 g1, i32x4 g2, i32x4 g3, i32x8 extra, i32 cpol))
typedef unsigned int u32x4 __attribute__((ext_vector_type(4)));
typedef int          i32x8 __attribute__((ext_vector_type(8)));
typedef int          i32x4 __attribute__((ext_vector_type(4)));

// Async copy: 4 bytes global -> LDS (per-lane addresses), tracked by ASYNCcnt.
__device__ __forceinline__ void async_g2l_b32(const float* g, float* l) {
#ifdef USE_ASYNC_BUILTIN
  __builtin_amdgcn_global_load_async_to_lds_b32((g_i32*)(g), (l_i32*)(l), 0, 0);
#else
  asm volatile("global_load_async_to_lds_b32 %0, %1, off"
               :
               : "v"((unsigned)(uintptr_t)(as3_void*)(l)), "v"(g)
               : "memory");
#endif
}

// Async copy: 16 bytes global -> LDS.
__device__ __forceinline__ void async_g2l_b128(const float* g, float* l) {
#ifdef USE_ASYNC_BUILTIN
  __builtin_amdgcn_global_load_async_to_lds_b128((g_v4i*)(g), (l_v4i*)(l), 0, 0);
#else
  asm volatile("global_load_async_to_lds_b128 %0, %1, off"
               :
               : "v"((unsigned)(uintptr_t)(as3_void*)(l)), "v"(g)
               : "memory");
#endif
}

__device__ __forceinline__ void wait_async0() {
#ifdef USE_WAITA_BUILTIN
  __builtin_amdgcn_s_wait_asynccnt(0);
#else
  asm volatile("s_wait_asynccnt 0" ::: "memory");
#endif
}

__device__ __forceinline__ void wait_tensor0() {
#ifdef USE_WAITT_BUILTIN
  __builtin_amdgcn_s_wait_tensorcnt(0);
#else
  asm volatile("s_wait_tensorcnt 0x0" ::: "memory");
#endif
}

// One block = one image n x one 16-filter tile. 256 threads (8 wave32s).
__global__ __launch_bounds__(NTHREADS) void adder2d_kernel(
    const float* __restrict__ x,   // (16, 64, 14, 14)
    const float* __restrict__ w,   // (64, 64, 3, 3) == (64, 576)
    float* __restrict__ out) {     // (16, 64, 14, 14)
  __shared__ __align__(16) float sX[C_][PADHW][PADHW];  // 64 KB zero-padded image
  __shared__ __align__(16) float sW[FT][D_];            // 36 KB weight tile

  const int tid = threadIdx.x;
  const int n   = blockIdx.x >> 2;
  const int f0  = (blockIdx.x & 3) * FT;

  // 1) Zero only the pad ring of sX (60 cells/channel; disjoint from interior,
  //    so no barrier needed before the async interior fill).
  for (int i = tid; i < C_ * 60; i += NTHREADS) {
    const int c = i / 60;
    const int r = i - c * 60;
    int row, col;
    if (r < 16)      { row = 0;          col = r;      }
    else if (r < 32) { row = PADHW - 1;  col = r - 16; }
    else if (r < 46) { row = r - 32 + 1; col = 0;      }
    else             { row = r - 46 + 1; col = PADHW - 1; }
    sX[c][row][col] = 0.0f;
  }

  // 2) Async-fill interior of sX from global x[n] (per-lane b32 async copies).
  {
    const float* xin = x + (size_t)n * C_ * P_;
    for (int i = tid; i < C_ * P_; i += NTHREADS) {
      const int c  = i / P_;
      const int r  = i - c * P_;
      const int h  = r / HW_;
      const int ww = r - h * HW_;
      async_g2l_b32(xin + i, &sX[c][h + 1][ww + 1]);
    }
  }

  // 3) Weight tile: one contiguous 36,864B region -> single TDM descriptor.
  const float* wbase = w + (size_t)f0 * D_;
#ifdef USE_TDM
  if (tid < 32) {  // wave 0 only (scalar branch; TDM ignores EXEC)
    const uint32_t elems    = FT * D_;  // 9216 fp32 elements, 1-D tile
    const uint32_t lds_base = (uint32_t)(uintptr_t)(as3_void*)&sW[0][0];
    const uint64_t gaddr    = (uint64_t)(uintptr_t)wbase;

    // D# group 0: count=1 (valid user desc), lds_addr, global_addr[56:0], type=2
    u32x4 g0;
    g0.x = 1u;
    g0.y = lds_base;
    g0.z = (uint32_t)gaddr;
    g0.w = (uint32_t)((gaddr >> 32) & 0x1FFFFFFu) | (2u << 30);

    // D# group 1: data_size=4B (code 2); tensor_dim0=tile_dim0=9216;
    // tensor_dim1=1, tile_dim1=1, tile_dim2=0; dim0 stride = 9216.
    i32x8 g1;
    g1[0] = (int)(2u << 16);                                   // data_size
    g1[1] = (int)((elems & 0xFFFFu) << 16);                    // tensor_dim0 lo16
    g1[2] = (int)((elems >> 16) | (1u << 16));                 // tensor_dim0 hi16 | tensor_dim1 lo16
    g1[3] = (int)(elems << 16);                                // tensor_dim1 hi16=0 | tile_dim0
    g1[4] = 1;                                                 // tile_dim1=1, tile_dim2=0
    g1[5] = (int)elems;                                        // tensor_dim0_stride lo32
    g1[6] = 0;                                                 // stride hi16 | dim1_stride lo16
    g1[7] = 0;

    // D# groups 2/3: benign values for unused higher dims.
    i32x4 g2; g2[0] = 1; g2[1] = 1; g2[2] = (int)elems; g2[3] = 0;
    i32x4 g3; g3[0] = 0; g3[1] = (int)(1u << 16); g3[2] = 0; g3[3] = 0;

    // Extra (non-architectural) 8-DWORD slot on the 6-arg toolchain: zero-fill.
    i32x8 gx = {0, 0, 0, 0, 0, 0, 0, 0};

    __builtin_amdgcn_tensor_load_to_lds(g0, g1, g2, g3, gx, 0);
  }
#else
  for (int i = tid; i < (FT * D_) / 4; i += NTHREADS) {
    const int fl = i / (D_ / 4);
    const int q  = i - fl * (D_ / 4);
    async_g2l_b128(wbase + fl * D_ + q * 4, &sW[fl][q * 4]);
  }
#endif

  wait_tensor0();   // no-op for waves with TENSORcnt==0
  wait_async0();
  __syncthreads();  // also orders the ring ds_stores

  // 4) Compute: thread <-> output position p; 16 filter accumulators in VGPRs.
  if (tid < P_) {
    const int ho = tid / HW_;
    const int wo = tid - ho * HW_;

    float acc[FT];
#pragma unroll
    for (int f = 0; f < FT; ++f) acc[f] = 0.0f;

    for (int c = 0; c < C_; ++c) {
#pragma unroll
      for (int kh = 0; kh < KS; ++kh) {
        // Lane-gather reads (consecutive p -> consecutive banks, conflict-free)
        const float x0 = sX[c][ho + kh][wo + 0];
        const float x1 = sX[c][ho + kh][wo + 1];
        const float x2 = sX[c][ho + kh][wo + 2];
        const int dbase = (c * KS + kh) * KS;
#pragma unroll
        for (int f = 0; f < FT; ++f) {
          // Broadcast reads (uniform across lanes -> wide ds loads)
          const float w0 = sW[f][dbase + 0];
          const float w1 = sW[f][dbase + 1];
          const float w2 = sW[f][dbase + 2];
          acc[f] += fabsf(w0 - x0) + fabsf(w1 - x1) + fabsf(w2 - x2);
        }
      }
    }

    float* o = out + ((size_t)n * F_ + f0) * P_ + tid;
#pragma unroll
    for (int f = 0; f < FT; ++f) o[f * P_] = -acc[f];
  }
}

extern "C" void kernel_launch(void* const* d_in, const int* in_sizes, int n_in,
                              void* d_out, int out_size, void* d_ws, size_t ws_size,
                              hipStream_t stream) {
  (void)in_sizes; (void)n_in; (void)d_ws; (void)ws_size; (void)out_size;
  const float* x = (const float*)d_in[0];  // (16,64,14,14) fp32
  const float* w = (const float*)d_in[1];  // (64,64,3,3)  fp32
  float* out = (float*)d_out;              // (16,64,14,14) fp32

  dim3 grid(N_ * (F_ / FT));  // 64 blocks
  adder2d_kernel<<<grid, NTHREADS, 0, stream>>>(x, w, out);
}